// MoEAudioClassifier_60284160967085
// MI455X (gfx1250) — compile-verified
//
#include <hip/hip_runtime.h>
#include <hip/hip_bf16.h>

// ---------------------------------------------------------------------------
// MoE audio classifier for MI455X (gfx1250), full-f32 via V_WMMA_F32_16X16X4_F32
// GEMM kernels use 4 N-tiles per wave sharing one A fetch (1.25 loads/wmma).
// ---------------------------------------------------------------------------

typedef float v2f __attribute__((ext_vector_type(2)));
typedef float v8f __attribute__((ext_vector_type(8)));

static __device__ __forceinline__ v8f wmma_f32(v2f a, v2f b, v8f c) {
  // D = A(16x4) * B(4x16) + C(16x16), f32 everywhere
  return __builtin_amdgcn_wmma_f32_16x16x4_f32(false, a, false, b, (short)0, c,
                                               false, false);
}

#define BATCH 2048
#define LSEQ  2048
#define CCH   128     // CONV channels
#define L1LEN 1024
#define L2LEN 512
#define DMODEL 1024
#define NEXP  6
#define HEXP  2048
#define NCLS  1000

// ===========================================================================
// Kernel 1: fused conv1(relu) + conv2(relu) + mean over length.
//   grid: BATCH * 16 blocks (each block: one batch row, 32 l2 positions)
//   conv1 tile recomputed into LDS, conv2 done as WMMA GEMM
//   M = co (128 -> 8 wave tiles), N = 32 l2 (2 n-tiles/wave), K = 128*5 = 640
// ===========================================================================
__global__ __launch_bounds__(256) void k_conv_fused(
    const float* __restrict__ x, const float* __restrict__ w1,
    const float* __restrict__ b1, const float* __restrict__ w2,
    const float* __restrict__ b2, float* __restrict__ hmean) {
  __shared__ float xs[144];         // x window for this tile
  __shared__ float h1s[CCH][68];    // conv1 outputs, j = l1 - (2*L0 - 2)

  const int b  = blockIdx.x >> 4;
  const int L0 = (blockIdx.x & 15) * 32;   // first l2 of this tile
  const int tid = threadIdx.x;

  // ---- load x window: xi in [4*L0-6, 4*L0+134], 141 values, zero-padded ----
  for (int j = tid; j < 141; j += 256) {
    int xi = 4 * L0 - 6 + j;
    xs[j] = (xi >= 0 && xi < LSEQ) ? x[b * LSEQ + xi] : 0.0f;
  }
  __syncthreads();

  // ---- conv1 + relu into LDS: l1 = 2*L0 - 2 + j, j in [0,67) ----
  for (int idx = tid; idx < CCH * 67; idx += 256) {
    int ci = idx / 67;
    int j  = idx % 67;
    int l1 = 2 * L0 - 2 + j;
    float v = 0.0f;
    if (l1 >= 0 && l1 < L1LEN) {
      float acc = b1[ci];
#pragma unroll
      for (int u = 0; u < 5; ++u) acc += xs[2 * j + u] * w1[ci * 5 + u];
      v = fmaxf(acc, 0.0f);
    }
    h1s[ci][j] = v;
  }
  __syncthreads();

  // ---- conv2 as WMMA GEMM ----
  const int lane = tid & 31;
  const int wave = tid >> 5;       // 8 waves -> 8 co tiles
  const int half = lane >> 4;
  const int lm   = lane & 15;
  const int mrow = wave * 16 + lm; // co for A loads

  v8f acc[2] = {};
  for (int k0 = 0; k0 < 640; k0 += 4) {
    const int kb = k0 + 2 * half;
    v2f a = *(const v2f*)(w2 + mrow * 640 + kb);   // w2[co][ci*5+t], 8B aligned
    const int ci0 = kb / 5,       t0 = kb - ci0 * 5;
    const int ci1 = (kb + 1) / 5, t1 = (kb + 1) - ci1 * 5;
#pragma unroll
    for (int nt = 0; nt < 2; ++nt) {
      const int nl = nt * 16 + lm;                 // local l2
      v2f bb;
      bb.x = h1s[ci0][2 * nl + t0];
      bb.y = h1s[ci1][2 * nl + t1];
      acc[nt] = wmma_f32(a, bb, acc[nt]);
    }
  }

  // ---- +bias, relu, partial mean, atomic accumulate ----
#pragma unroll
  for (int v = 0; v < 8; ++v) {
    const int co = wave * 16 + v + 8 * half;
    const float bias = b2[co];
    float t = fmaxf(acc[0][v] + bias, 0.0f) + fmaxf(acc[1][v] + bias, 0.0f);
#pragma unroll
    for (int off = 1; off < 16; off <<= 1) t += __shfl_xor(t, off, 32);
    if (lm == 0) atomicAdd(&hmean[b * CCH + co], t * (1.0f / (float)L2LEN));
  }
}

// ===========================================================================
// Kernel 2: proj GEMM  h = hmean(2048x128) @ proj_w^T(128x1024) + proj_b
//   wave computes 4 N-tiles (64 cols) sharing the A fetch.
//   tiles: 128 b-tiles x 16 n-groups -> 2048 waves -> 256 blocks
// ===========================================================================
__global__ __launch_bounds__(256) void k_proj(const float* __restrict__ hm,
                                              const float* __restrict__ pw,
                                              const float* __restrict__ pb,
                                              float* __restrict__ h) {
  const int lane = threadIdx.x & 31, wave = threadIdx.x >> 5;
  const int t   = blockIdx.x * 8 + wave;   // 2048 wave tiles
  const int bt  = t >> 4;
  const int ntg = t & 15;
  const int half = lane >> 4, lm = lane & 15;
  const int mrow  = bt * 16 + lm;
  const int ncol0 = ntg * 64 + lm;

  v8f c[4] = {};
  for (int k0 = 0; k0 < CCH; k0 += 4) {
    const int kb = k0 + 2 * half;
    v2f a = *(const v2f*)(hm + mrow * CCH + kb);
#pragma unroll
    for (int nt = 0; nt < 4; ++nt) {
      v2f bb = *(const v2f*)(pw + (ncol0 + nt * 16) * CCH + kb);
      c[nt] = wmma_f32(a, bb, c[nt]);
    }
  }
#pragma unroll
  for (int nt = 0; nt < 4; ++nt) {
    const int ncol = ncol0 + nt * 16;
    const float bias = pb[ncol];
#pragma unroll
    for (int v = 0; v < 8; ++v) {
      const int row = bt * 16 + v + 8 * half;
      h[row * DMODEL + ncol] = c[nt][v] + bias;
    }
  }
}

// ===========================================================================
// Kernel 3: router + softmax + top-2 -> combine (2048 x 6); one wave per row
// ===========================================================================
__global__ __launch_bounds__(256) void k_router(const float* __restrict__ h,
                                                const float* __restrict__ rw,
                                                const float* __restrict__ rb,
                                                float* __restrict__ combine) {
  const int lane = threadIdx.x & 31, wave = threadIdx.x >> 5;
  const int b = blockIdx.x * 8 + wave;
  float l[NEXP];
#pragma unroll
  for (int e = 0; e < NEXP; ++e) {
    float s = 0.0f;
    for (int d = lane; d < DMODEL; d += 32)
      s += h[b * DMODEL + d] * rw[e * DMODEL + d];
#pragma unroll
    for (int off = 1; off < 32; off <<= 1) s += __shfl_xor(s, off, 32);
    l[e] = s + rb[e];
  }
  float mx = l[0];
#pragma unroll
  for (int e = 1; e < NEXP; ++e) mx = fmaxf(mx, l[e]);
  float p[NEXP], se = 0.0f;
#pragma unroll
  for (int e = 0; e < NEXP; ++e) { p[e] = __expf(l[e] - mx); se += p[e]; }
  const float inv = 1.0f / se;
#pragma unroll
  for (int e = 0; e < NEXP; ++e) p[e] *= inv;
  int i0 = 0;
#pragma unroll
  for (int e = 1; e < NEXP; ++e) if (p[e] > p[i0]) i0 = e;
  int i1 = (i0 == 0) ? 1 : 0;
#pragma unroll
  for (int e = 0; e < NEXP; ++e)
    if (e != i0 && e != i1 && p[e] > p[i1]) i1 = e;
  if (lane < NEXP)
    combine[b * NEXP + lane] =
        (lane == i0) ? p[i0] : ((lane == i1) ? p[i1] : 0.0f);
}

// ===========================================================================
// Kernel 4: expert layer 1 (dense over all experts)
//   hid[b,e,h'] = relu(h[b,:] . exp_w1[e,h',:] + exp_b1[e,h'])
//   wave computes 4 ht-tiles (64 cols) sharing A fetch.
//   tiles: E(6) x 128 b-tiles x 32 ht-groups -> 24576 waves -> 3072 blocks
// ===========================================================================
__global__ __launch_bounds__(256) void k_exp1(const float* __restrict__ h,
                                              const float* __restrict__ w1,
                                              const float* __restrict__ b1,
                                              float* __restrict__ hid) {
  const int lane = threadIdx.x & 31, wave = threadIdx.x >> 5;
  const int t   = blockIdx.x * 8 + wave;   // 24576 wave tiles
  const int e   = t / 4096;
  const int r   = t - e * 4096;
  const int bt  = r >> 5;
  const int htg = r & 31;
  const int half = lane >> 4, lm = lane & 15;
  const int mrow  = bt * 16 + lm;        // batch row
  const int ncol0 = htg * 64 + lm;       // first hidden unit

  const float* arow = h + (long)mrow * DMODEL;
  const float* w1e  = w1 + (long)e * HEXP * DMODEL;

  v8f c[4] = {};
  for (int k0 = 0; k0 < DMODEL; k0 += 4) {
    const int kb = k0 + 2 * half;
    v2f a = *(const v2f*)(arow + kb);
#pragma unroll
    for (int nt = 0; nt < 4; ++nt) {
      v2f bb = *(const v2f*)(w1e + (long)(ncol0 + nt * 16) * DMODEL + kb);
      c[nt] = wmma_f32(a, bb, c[nt]);
    }
  }
#pragma unroll
  for (int nt = 0; nt < 4; ++nt) {
    const int ncol = ncol0 + nt * 16;
    const float bias = b1[e * HEXP + ncol];
#pragma unroll
    for (int v = 0; v < 8; ++v) {
      const int row = bt * 16 + v + 8 * half;
      hid[((long)row * NEXP + e) * HEXP + ncol] = fmaxf(c[nt][v] + bias, 0.0f);
    }
  }
}

// ===========================================================================
// Kernel 5: expert layer 2 + combine-weighted accumulation into moe
//   moe[b,d] = sum_e combine[b,e]*(hid[b,e,:] . exp_w2[e,d,:] + exp_b2[e,d])
//   wave computes 4 d-tiles (64 cols); experts looped, A fetch shared.
//   tiles: 128 b-tiles x 16 d-groups -> 2048 waves -> 256 blocks
// ===========================================================================
__global__ __launch_bounds__(256) void k_exp2_moe(
    const float* __restrict__ hid, const float* __restrict__ w2,
    const float* __restrict__ b2, const float* __restrict__ combine,
    float* __restrict__ moe) {
  const int lane = threadIdx.x & 31, wave = threadIdx.x >> 5;
  const int t   = blockIdx.x * 8 + wave;   // 2048 wave tiles
  const int bt  = t >> 4;
  const int dtg = t & 15;
  const int half = lane >> 4, lm = lane & 15;
  const int mrow  = bt * 16 + lm;   // batch row
  const int ncol0 = dtg * 64 + lm;  // first d index

  float out[4][8] = {};
  for (int e = 0; e < NEXP; ++e) {
    const float* arow = hid + ((long)mrow * NEXP + e) * HEXP;
    const float* w2e  = w2 + (long)e * DMODEL * HEXP;
    v8f c[4] = {};
    for (int k0 = 0; k0 < HEXP; k0 += 4) {
      const int kb = k0 + 2 * half;
      v2f a = *(const v2f*)(arow + kb);
#pragma unroll
      for (int nt = 0; nt < 4; ++nt) {
        v2f bb = *(const v2f*)(w2e + (long)(ncol0 + nt * 16) * HEXP + kb);
        c[nt] = wmma_f32(a, bb, c[nt]);
      }
    }
#pragma unroll
    for (int nt = 0; nt < 4; ++nt) {
      const int ncol = ncol0 + nt * 16;
      const float bias = b2[e * DMODEL + ncol];
#pragma unroll
      for (int v = 0; v < 8; ++v) {
        const int row = bt * 16 + v + 8 * half;
        out[nt][v] += combine[row * NEXP + e] * (c[nt][v] + bias);
      }
    }
  }
#pragma unroll
  for (int nt = 0; nt < 4; ++nt) {
    const int ncol = ncol0 + nt * 16;
#pragma unroll
    for (int v = 0; v < 8; ++v) {
      const int row = bt * 16 + v + 8 * half;
      moe[row * DMODEL + ncol] = out[nt][v];
    }
  }
}

// ===========================================================================
// Kernel 6: classifier  out = moe(2048x1024) @ cls_w^T(1024x1000) + cls_b
//   63 N-tiles = 21 groups of 3; loads clamped, stores guarded per tile.
//   tiles: 128 b-tiles x 21 n-groups -> 2688 waves -> 336 blocks
// ===========================================================================
__global__ __launch_bounds__(256) void k_cls(const float* __restrict__ moe,
                                             const float* __restrict__ cw,
                                             const float* __restrict__ cb,
                                             float* __restrict__ out) {
  const int lane = threadIdx.x & 31, wave = threadIdx.x >> 5;
  const int t   = blockIdx.x * 8 + wave;   // 2688 wave tiles
  const int bt  = t / 21;
  const int ntg = t - bt * 21;
  const int half = lane >> 4, lm = lane & 15;
  const int mrow  = bt * 16 + lm;
  const int ncol0 = ntg * 48 + lm;

  int nclamp[3];
#pragma unroll
  for (int nt = 0; nt < 3; ++nt) {
    const int ncol = ncol0 + nt * 16;
    nclamp[nt] = (ncol < NCLS) ? ncol : (NCLS - 1);
  }

  const float* arow = moe + (long)mrow * DMODEL;
  v8f c[3] = {};
  for (int k0 = 0; k0 < DMODEL; k0 += 4) {
    const int kb = k0 + 2 * half;
    v2f a = *(const v2f*)(arow + kb);
#pragma unroll
    for (int nt = 0; nt < 3; ++nt) {
      v2f bb = *(const v2f*)(cw + (long)nclamp[nt] * DMODEL + kb);
      c[nt] = wmma_f32(a, bb, c[nt]);
    }
  }
#pragma unroll
  for (int nt = 0; nt < 3; ++nt) {
    const int ncol = ncol0 + nt * 16;
    if (ncol < NCLS) {
      const float bias = cb[ncol];
#pragma unroll
      for (int v = 0; v < 8; ++v) {
        const int row = bt * 16 + v + 8 * half;
        out[row * NCLS + ncol] = c[nt][v] + bias;
      }
    }
  }
}

// ===========================================================================
extern "C" void kernel_launch(void* const* d_in, const int* in_sizes, int n_in,
                              void* d_out, int out_size, void* d_ws,
                              size_t ws_size, hipStream_t stream) {
  const float* x       = (const float*)d_in[0];
  const float* conv1_w = (const float*)d_in[1];
  const float* conv1_b = (const float*)d_in[2];
  const float* conv2_w = (const float*)d_in[3];
  const float* conv2_b = (const float*)d_in[4];
  const float* proj_w  = (const float*)d_in[5];
  const float* proj_b  = (const float*)d_in[6];
  const float* rout_w  = (const float*)d_in[7];
  const float* rout_b  = (const float*)d_in[8];
  const float* exp_w1  = (const float*)d_in[9];
  const float* exp_b1  = (const float*)d_in[10];
  const float* exp_w2  = (const float*)d_in[11];
  const float* exp_b2  = (const float*)d_in[12];
  const float* cls_w   = (const float*)d_in[13];
  const float* cls_b   = (const float*)d_in[14];
  float* out = (float*)d_out;

  // workspace layout (floats)
  float* ws      = (float*)d_ws;
  float* hmean   = ws;                              // 2048*128
  float* h       = hmean + (size_t)BATCH * CCH;     // 2048*1024
  float* combine = h + (size_t)BATCH * DMODEL;      // 2048*6
  float* hid     = combine + (size_t)BATCH * NEXP;  // 2048*6*2048
  float* moe     = hid + (size_t)BATCH * NEXP * HEXP;  // 2048*1024

  // hmean is accumulated with atomics -> zero it (capturable memset)
  hipMemsetAsync(hmean, 0, (size_t)BATCH * CCH * sizeof(float), stream);

  k_conv_fused<<<BATCH * 16, 256, 0, stream>>>(x, conv1_w, conv1_b, conv2_w,
                                               conv2_b, hmean);
  k_proj<<<256, 256, 0, stream>>>(hmean, proj_w, proj_b, h);
  k_router<<<256, 256, 0, stream>>>(h, rout_w, rout_b, combine);
  k_exp1<<<3072, 256, 0, stream>>>(h, exp_w1, exp_b1, hid);
  k_exp2_moe<<<256, 256, 0, stream>>>(hid, exp_w2, exp_b2, combine, moe);
  k_cls<<<336, 256, 0, stream>>>(moe, cls_w, cls_b, out);
}